// MAMBA_80144089744079
// MI455X (gfx1250) — compile-verified
//
#include <hip/hip_runtime.h>
#include <math.h>

typedef float v2f __attribute__((ext_vector_type(2)));
typedef float v8f __attribute__((ext_vector_type(8)));

#define BSZ 8
#define EMB 256
#define OC  64
#define DI  128
#define NST 16
#define RNK 4
#define KD  4
#define HH  64
#define WW  64
#define LSZ 4096
#define BL  (BSZ*LSZ)   /* 32768 rows (b,l) */

// ---------------------------------------------------------------------------
// fp32 WMMA GEMM: out[M,N] = A[M,KTOT] * B[KTOT,N]  (+scale/bias/resid)
// Compile-time specialized:
//   ANCHW : A element (m,k) at A[(m>>12)*KTOT*4096 + k*4096 + (m&4095)]
//   BTRANS: B element (k,n) at Bw[n*ldb + k]
//   KTOT  : full K extent -> fully unrolled branch-free inner loop
//   BMASK : apply branch-free edge masking for N not multiple of 16
// Each wave computes a 16x16 tile with v_wmma_f32_16x16x4_f32.
// ---------------------------------------------------------------------------
template<int ANCHW, int BTRANS, int KTOT, int BMASK>
__global__ __launch_bounds__(256) void gemm_wmma(
    const float* __restrict__ A, const float* __restrict__ Bw,
    float* __restrict__ C,
    int N, int lda, int ldb, int ldc,
    const float* __restrict__ scale, const float* __restrict__ bias,
    const float* __restrict__ resid, int row_remap)
{
  int lane = threadIdx.x & 31;
  int wave = threadIdx.x >> 5;
  int m0 = (blockIdx.x * 8 + wave) * 16;
  int n0 = blockIdx.y * 16;
  int r    = lane & 15;
  int koff = (lane >> 4) << 1;          // 0 for lanes 0-15, 2 for lanes 16-31
  int mrow = m0 + r;
  int ncol = n0 + r;

  // loop-invariant fragment pointers
  const float* ap;
  const float* bp;
  float bmask = 1.f;
  if (ANCHW) {
    ap = A + (size_t)(mrow >> 12) * ((size_t)KTOT * LSZ) + (mrow & 4095)
           + (size_t)koff * LSZ;
  } else {
    ap = A + (size_t)mrow * lda + koff;
  }
  if (BTRANS) {
    int nc = ncol;
    if (BMASK) {
      nc = ncol < N ? ncol : (N - 1);
      bmask = ncol < N ? 1.f : 0.f;
    }
    bp = Bw + (size_t)nc * ldb + koff;
  } else {
    bp = Bw + (size_t)koff * ldb + ncol;
  }

  v8f acc = {0.f,0.f,0.f,0.f,0.f,0.f,0.f,0.f};
#pragma unroll
  for (int kb = 0; kb < KTOT; kb += 4) {
    v2f a, b;
    if (ANCHW) {                        // k-stride = 4096 floats
      a.x = ap[0];
      a.y = ap[LSZ];
      ap += (size_t)4 * LSZ;
    } else {                            // contiguous k: one 8B load
      a = *(const v2f*)ap;
      ap += 4;
    }
    if (BTRANS) {                       // contiguous k: one 8B load
      b = *(const v2f*)bp;
      if (BMASK) { b.x *= bmask; b.y *= bmask; }
      bp += 4;
    } else {                            // k-stride = ldb
      b.x = bp[0];
      b.y = bp[ldb];
      bp += (size_t)4 * ldb;
    }
    acc = __builtin_amdgcn_wmma_f32_16x16x4_f32(false, a, false, b,
                                                (short)0, acc, false, false);
  }

  int rbase = m0 + ((lane >> 4) << 3);  // rows 0-7 (lanes<16) or 8-15
  int cc = n0 + (lane & 15);
  if (cc < N) {
    float sc = scale ? scale[cc] : 1.f;
    float bi = bias  ? bias[cc]  : 0.f;
#pragma unroll
    for (int i = 0; i < 8; ++i) {
      int rr = rbase + i;
      int srow = rr;
      if (row_remap) {
        int bb = rr >> 12, hw = rr & 4095;
        srow = (bb << 12) | ((hw & 63) << 6) | (hw >> 6);
      }
      float v = acc[i] * sc + bi;
      if (resid) v += resid[(size_t)srow * ldc + cc];
      C[(size_t)srow * ldc + cc] = v;
    }
  }
}

// ---------------------------------------------------------------------------
// Direct KSxKS conv (OC->OC) on channel-last input (b, h*64+w, ic), BN-folded
// scale/bias, output written spatially transposed: (b, w*64+h, oc).
// ---------------------------------------------------------------------------
template<int KS, int PAD>
__global__ __launch_bounds__(256) void conv_kxk(
    const float* __restrict__ in, const float* __restrict__ w,
    const float* __restrict__ s, const float* __restrict__ bias,
    float* __restrict__ out)
{
  int idx = blockIdx.x * 256 + threadIdx.x;
  int b   = idx >> 18;
  int rem = idx & ((1 << 18) - 1);
  int hw  = rem >> 6;
  int oc  = rem & 63;
  int h = hw >> 6, ww_ = hw & 63;
  float acc = 0.f;
#pragma unroll
  for (int ky = 0; ky < KS; ++ky) {
    int ih = h + ky - PAD;
    if (ih < 0 || ih >= HH) continue;
#pragma unroll
    for (int kx = 0; kx < KS; ++kx) {
      int iw = ww_ + kx - PAD;
      if (iw < 0 || iw >= WW) continue;
      const float* ip = in + ((size_t)b * LSZ + ih * 64 + iw) * OC;
      const float* wp = w + (size_t)oc * OC * KS * KS + ky * KS + kx;
      for (int ic = 0; ic < OC; ++ic)
        acc += ip[ic] * wp[(size_t)ic * KS * KS];
    }
  }
  out[((size_t)b * LSZ + (ww_ << 6) + h) * OC + oc] = acc * s[oc] + bias[oc];
}

// LayerNorm over 64 channels (one wave per position, 2 ch/lane)
__global__ __launch_bounds__(256) void ln64_kernel(
    const float* __restrict__ x, const float* __restrict__ g,
    const float* __restrict__ b, float* __restrict__ out)
{
  int lane = threadIdx.x & 31;
  int p = blockIdx.x * 8 + (threadIdx.x >> 5);
  const float* xp = x + (size_t)p * 64;
  float x0 = xp[lane], x1 = xp[lane + 32];
  float s = x0 + x1, q = x0 * x0 + x1 * x1;
  for (int off = 16; off > 0; off >>= 1) {
    s += __shfl_xor(s, off, 32);
    q += __shfl_xor(q, off, 32);
  }
  float mean = s * (1.f / 64.f);
  float var  = q * (1.f / 64.f) - mean * mean;
  float rstd = rsqrtf(var + 1e-5f);
  float* op = out + (size_t)p * 64;
  op[lane]      = (x0 - mean) * rstd * g[lane]      + b[lane];
  op[lane + 32] = (x1 - mean) * rstd * g[lane + 32] + b[lane + 32];
}

// Depthwise 3x3 + bias + SiLU on xc (first DI cols of xz), scattered into the
// four cross-scan orders, stored position-major: xs[k][b][l][d].
__global__ __launch_bounds__(256) void dwconv_scatter(
    const float* __restrict__ xz, const float* __restrict__ cw,
    const float* __restrict__ cb, float* __restrict__ xs)
{
  int idx = blockIdx.x * 256 + threadIdx.x;
  int b   = idx >> 19;
  int rem = idx & ((1 << 19) - 1);
  int l = rem >> 7;
  int d = rem & 127;
  int hs = l >> 6, ws = l & 63;
  float acc = cb[d];
#pragma unroll
  for (int ky = 0; ky < 3; ++ky) {
    int ih = hs + ky - 1;
    if (ih < 0 || ih >= 64) continue;
#pragma unroll
    for (int kx = 0; kx < 3; ++kx) {
      int iw = ws + kx - 1;
      if (iw < 0 || iw >= 64) continue;
      acc += xz[((size_t)b * LSZ + ih * 64 + iw) * 256 + d] * cw[d * 9 + ky * 3 + kx];
    }
  }
  float v = acc / (1.f + __expf(-acc));   // SiLU
  size_t S = (size_t)BL * DI;
  int l1 = (ws << 6) | hs;                // wh order index
  size_t base = (size_t)b * LSZ;
  xs[0 * S + (base + l) * DI + d]               = v;
  xs[1 * S + (base + l1) * DI + d]              = v;
  xs[2 * S + (base + (LSZ - 1 - l)) * DI + d]   = v;
  xs[3 * S + (base + (LSZ - 1 - l1)) * DI + d]  = v;
}

// Selective scan: one workgroup per (k,b); thread owns (d, 8 states).
// dt projection (RNK=4) fused; B/C shared across lanes via wave32 shuffles.
__global__ __launch_bounds__(256) void scan_kernel(
    const float* __restrict__ xdbl, const float* __restrict__ xs,
    const float* __restrict__ dtw, const float* __restrict__ dtb,
    const float* __restrict__ Alog, const float* __restrict__ Dp,
    float* __restrict__ ys)
{
  int k = blockIdx.x >> 3;
  int b = blockIdx.x & 7;
  int tid  = threadIdx.x;
  int lane = tid & 31;
  int d    = tid >> 1;
  int half = tid & 1;
  int nb   = half * 8;

  float Av[8], hreg[8];
#pragma unroll
  for (int j = 0; j < 8; ++j) {
    Av[j]   = -__expf(Alog[((k * DI) + d) * NST + nb + j]);
    hreg[j] = 0.f;
  }
  float w0 = dtw[((k * DI) + d) * RNK + 0];
  float w1 = dtw[((k * DI) + d) * RNK + 1];
  float w2 = dtw[((k * DI) + d) * RNK + 2];
  float w3 = dtw[((k * DI) + d) * RNK + 3];
  float dtbd = dtb[k * DI + d];
  float Dv   = Dp[k * DI + d];

  const float* xdk = xdbl + ((size_t)k * BL + (size_t)b * LSZ) * 36;
  const float* xsk = xs   + ((size_t)k * BL + (size_t)b * LSZ) * DI;
  float*       ysk = ys   + ((size_t)k * BL + (size_t)b * LSZ) * DI;

  for (int l = 0; l < LSZ; ++l) {
    const float* xp = xdk + (size_t)l * 36;
    __builtin_prefetch(xp + 36, 0, 1);            // next step's projection row
    float bcv = xp[4 + lane];                     // lanes 0-15: B, 16-31: C
    float d0 = xp[0], d1 = xp[1], d2 = xp[2], d3 = xp[3];  // uniform -> s_load
    float u = xsk[(size_t)l * DI + d];
    float raw = dtbd + w0 * d0 + w1 * d1 + w2 * d2 + w3 * d3;
    float dtv = (raw > 20.f) ? raw : __logf(1.f + __expf(raw));  // softplus
    float acc = 0.f;
#pragma unroll
    for (int j = 0; j < 8; ++j) {
      float Bn = __shfl(bcv, nb + j, 32);
      float Cn = __shfl(bcv, 16 + nb + j, 32);
      float dA = __expf(dtv * Av[j]);
      hreg[j] = dA * hreg[j] + (dtv * Bn) * u;
      acc += hreg[j] * Cn;
    }
    acc += __shfl_xor(acc, 1, 32);
    if (half == 0) ysk[(size_t)l * DI + d] = acc + Dv * u;
  }
}

// Merge 4 scan directions back to hw order + LayerNorm(128) + SiLU(z) gate.
__global__ __launch_bounds__(256) void merge_ln_gate(
    const float* __restrict__ ys, const float* __restrict__ xz,
    const float* __restrict__ g, const float* __restrict__ bb,
    float* __restrict__ out)
{
  int lane = threadIdx.x & 31;
  int p = blockIdx.x * 8 + (threadIdx.x >> 5);
  int b = p >> 12;
  int l = p & 4095;
  int l1 = ((l & 63) << 6) | (l >> 6);
  size_t S = (size_t)BL * DI;
  const float* r0 = ys + 0 * S + ((size_t)b * LSZ + l) * DI;
  const float* r1 = ys + 1 * S + ((size_t)b * LSZ + l1) * DI;
  const float* r2 = ys + 2 * S + ((size_t)b * LSZ + (LSZ - 1 - l)) * DI;
  const float* r3 = ys + 3 * S + ((size_t)b * LSZ + (LSZ - 1 - l1)) * DI;
  float yv[4]; float s = 0.f, q = 0.f;
#pragma unroll
  for (int j = 0; j < 4; ++j) {
    int d = lane + 32 * j;
    float v = r0[d] + r1[d] + r2[d] + r3[d];
    yv[j] = v; s += v; q += v * v;
  }
  for (int off = 16; off > 0; off >>= 1) {
    s += __shfl_xor(s, off, 32);
    q += __shfl_xor(q, off, 32);
  }
  float mean = s * (1.f / 128.f);
  float var  = q * (1.f / 128.f) - mean * mean;
  float rstd = rsqrtf(var + 1e-5f);
  const float* zp = xz + (size_t)p * 256 + 128;
  float* op = out + (size_t)p * DI;
#pragma unroll
  for (int j = 0; j < 4; ++j) {
    int d = lane + 32 * j;
    float z = zp[d];
    float sz = z / (1.f + __expf(-z));
    op[d] = ((yv[j] - mean) * rstd * g[d] + bb[d]) * sz;
  }
}

// Concat (along W-axis) + transpose + bilinear resize 192->256 + reshape +
// residual add + ReLU. Output flat layout == (B,256,64,64).
__global__ __launch_bounds__(256) void final_resize_relu(
    const float* __restrict__ v0, const float* __restrict__ v1,
    const float* __restrict__ v2, const float* __restrict__ x,
    float* __restrict__ out)
{
  int idx = blockIdx.x * 256 + threadIdx.x;
  int b    = idx >> 20;
  int rem  = idx & ((1 << 20) - 1);
  int oc   = rem >> 14;
  int rem2 = rem & 16383;
  int h = rem2 >> 8;
  int j = rem2 & 255;
  float sx = (j + 0.5f) * 0.75f - 0.5f;    // half-pixel centers, 256->192
  float fl = floorf(sx);
  float f  = sx - fl;
  int j0 = (int)fl;
  int ja = min(max(j0, 0), 191);
  int jb = min(max(j0 + 1, 0), 191);
  auto gv = [&](int jj) {
    int br = jj >> 6;
    const float* v = (br == 0) ? v0 : ((br == 1) ? v1 : v2);
    return v[((size_t)b * LSZ + ((jj & 63) << 6) + h) * OC + oc];
  };
  float val = (1.f - f) * gv(ja) + f * gv(jb);
  out[idx] = fmaxf(val + x[idx], 0.f);
}

// ---------------------------------------------------------------------------
extern "C" void kernel_launch(void* const* d_in, const int* in_sizes, int n_in,
                              void* d_out, int out_size, void* d_ws, size_t ws_size,
                              hipStream_t stream)
{
  (void)in_sizes; (void)n_in; (void)out_size; (void)ws_size;
  const float* X = (const float*)d_in[0];
  float* ws = (float*)d_ws;

  const size_t TBL = (size_t)BL * OC;          // 2,097,152 floats
  float* t[3]   = { ws + 0 * TBL, ws + 1 * TBL, ws + 2 * TBL };
  float* vss[3] = { ws + 3 * TBL, ws + 4 * TBL, ws + 5 * TBL };
  float* c1buf  = ws + 6 * TBL;
  float* lnbuf  = ws + 7 * TBL;
  float* xz     = ws + 8 * TBL;                // BL*256
  float* xs     = xz   + (size_t)BL * 256;     // 4*BL*128
  float* xdbl   = xs   + (size_t)4 * BL * DI;  // 4*BL*36
  float* ysb    = xdbl + (size_t)4 * BL * 36;  // 4*BL*128
  float* ygate  = ysb  + (size_t)4 * BL * DI;  // BL*128

  const dim3 blk(256);
  const dim3 gconv(BL / 128, OC / 16);         // conv1x1 / out_proj: N=64
  const dim3 ginpj(BL / 128, (2 * DI) / 16);   // in_proj: N=256
  const dim3 gxdbl(BL / 128, 3);               // xdbl: N=36 -> 3 tiles

  // --- branch convs (BasicConv2d: conv * s + b, no relu) ---
  // b0: 1x1 -> t[0], spatially transposed store
  gemm_wmma<1,1,EMB,0><<<gconv, blk, 0, stream>>>(X, (const float*)d_in[1], t[0],
      OC, 0, EMB, OC, (const float*)d_in[2], (const float*)d_in[3], nullptr, 1);
  // b1: 1x1 -> c1buf (natural order), then 3x3 -> t[1]
  gemm_wmma<1,1,EMB,0><<<gconv, blk, 0, stream>>>(X, (const float*)d_in[4], c1buf,
      OC, 0, EMB, OC, (const float*)d_in[5], (const float*)d_in[6], nullptr, 0);
  conv_kxk<3,1><<<(int)(TBL / 256), blk, 0, stream>>>(c1buf,
      (const float*)d_in[7], (const float*)d_in[8], (const float*)d_in[9], t[1]);
  // b2: 1x1 -> c1buf, then 5x5 -> t[2]
  gemm_wmma<1,1,EMB,0><<<gconv, blk, 0, stream>>>(X, (const float*)d_in[10], c1buf,
      OC, 0, EMB, OC, (const float*)d_in[11], (const float*)d_in[12], nullptr, 0);
  conv_kxk<5,2><<<(int)(TBL / 256), blk, 0, stream>>>(c1buf,
      (const float*)d_in[13], (const float*)d_in[14], (const float*)d_in[15], t[2]);

  // --- VSS blocks (sequential per branch, SS2D workspace reused) ---
  const int rbase[3] = {16, 29, 42};
  for (int r = 0; r < 3; ++r) {
    const float* ln_g   = (const float*)d_in[rbase[r] + 0];
    const float* ln_b   = (const float*)d_in[rbase[r] + 1];
    const float* in_w   = (const float*)d_in[rbase[r] + 2];
    const float* conv_w = (const float*)d_in[rbase[r] + 3];
    const float* conv_b = (const float*)d_in[rbase[r] + 4];
    const float* xproj  = (const float*)d_in[rbase[r] + 5];
    const float* dtw    = (const float*)d_in[rbase[r] + 6];
    const float* dtb    = (const float*)d_in[rbase[r] + 7];
    const float* Alog   = (const float*)d_in[rbase[r] + 8];
    const float* Dp     = (const float*)d_in[rbase[r] + 9];
    const float* on_g   = (const float*)d_in[rbase[r] + 10];
    const float* on_b   = (const float*)d_in[rbase[r] + 11];
    const float* out_w  = (const float*)d_in[rbase[r] + 12];

    ln64_kernel<<<BL / 8, blk, 0, stream>>>(t[r], ln_g, ln_b, lnbuf);
    // in_proj: (BL x 64) @ (64 x 256)
    gemm_wmma<0,0,OC,0><<<ginpj, blk, 0, stream>>>(lnbuf, in_w, xz,
        2 * DI, OC, 2 * DI, 2 * DI, nullptr, nullptr, nullptr, 0);
    dwconv_scatter<<<(BL * DI) / 256, blk, 0, stream>>>(xz, conv_w, conv_b, xs);
    // x_dbl per direction: (BL x 128) @ (128 x 36), B transposed, edge-masked
    for (int k = 0; k < KD; ++k)
      gemm_wmma<0,1,DI,1><<<gxdbl, blk, 0, stream>>>(
          xs + (size_t)k * BL * DI, xproj + (size_t)k * 36 * DI,
          xdbl + (size_t)k * BL * 36, 36, DI, DI, 36,
          nullptr, nullptr, nullptr, 0);
    scan_kernel<<<KD * BSZ, blk, 0, stream>>>(xdbl, xs, dtw, dtb, Alog, Dp, ysb);
    merge_ln_gate<<<BL / 8, blk, 0, stream>>>(ysb, xz, on_g, on_b, ygate);
    // out_proj: (BL x 128) @ (128 x 64) + residual t[r]
    gemm_wmma<0,0,DI,0><<<gconv, blk, 0, stream>>>(ygate, out_w, vss[r],
        OC, DI, OC, OC, nullptr, nullptr, t[r], 0);
  }

  // --- concat + resize + residual + relu ---
  final_resize_relu<<<(8 * 64 * 64 * 256) / 256, blk, 0, stream>>>(
      vss[0], vss[1], vss[2], X, (float*)d_out);
}